// InitialTSDFModel_7352984010838
// MI455X (gfx1250) — compile-verified
//
#include <hip/hip_runtime.h>

#define BIGF 1.0e10f

typedef __attribute__((ext_vector_type(2))) float v2f;
typedef __attribute__((ext_vector_type(8))) float v8f;

// ---------------------------------------------------------------------------
// Kernel 1: interface init  -> d0 (frozen <=> d0 < BIG), and dA = d0
// ---------------------------------------------------------------------------
__global__ __launch_bounds__(256) void init_kernel(const float* __restrict__ phi,
                                                   float* __restrict__ d0,
                                                   float* __restrict__ dA) {
    int i = blockIdx.x * 256 + threadIdx.x;           // 0..262143
    int y = i >> 9, x = i & 511;
    float c  = phi[i];
    float up = phi[max(y - 1, 0)   * 512 + x];
    float dn = phi[min(y + 1, 511) * 512 + x];
    float lf = phi[y * 512 + max(x - 1, 0)];
    float rt = phi[y * 512 + min(x + 1, 511)];

    auto dk = [&](float nb) -> float {
        bool cross = (c * nb) < 0.0f;
        return cross ? fabsf(c) / fmaxf(fabsf(c - nb), 1e-12f) : BIGF;
    };
    float da = fminf(dk(up), dk(dn));
    float db = fminf(dk(lf), dk(rt));
    float inv2 = (da < BIGF ? 1.0f / (da * da) : 0.0f)
               + (db < BIGF ? 1.0f / (db * db) : 0.0f);
    bool frozen = (inv2 > 0.0f) || (c == 0.0f);
    float v = frozen ? 1.0f / sqrtf(fmaxf(inv2, 1e-20f)) : BIGF;
    v = (c == 0.0f) ? 0.0f : v;
    d0[i] = v;
    dA[i] = v;
}

// ---------------------------------------------------------------------------
// Kernel 2: temporally-blocked eikonal sweeps.
// Tile: 32x32 interior + 8 halo each side = 48x48 staged, stored in 50x50 LDS
// with a permanent BIG border ring. nsweeps <= 8 keeps the interior exact
// (wrong-halo contamination advances 1 cell/sweep; it is also one-sided >=
// truth because beyond-halo neighbors read BIG).
// ---------------------------------------------------------------------------
__global__ __launch_bounds__(256) void sweep_kernel(const float* __restrict__ din,
                                                    float* __restrict__ dout,
                                                    const float* __restrict__ d0,
                                                    int nsweeps) {
    __shared__ float s0[2500];
    __shared__ float sA[2500];
    __shared__ float sB[2500];

    int t  = threadIdx.x;
    int TI = blockIdx.y, TJ = blockIdx.x;
    int gy0 = TI * 32 - 9;   // global row of staged local row ly  is gy0+ly
    int gx0 = TJ * 32 - 9;

    // Stage 48x48 (ly,lx in [1,48]) into both buffers; border + out-of-grid = BIG.
    for (int idx = t; idx < 2500; idx += 256) {
        int ly = idx / 50, lx = idx % 50;
        int gy = gy0 + ly, gx = gx0 + lx;
        float v = BIGF, z = BIGF;
        if (ly >= 1 && ly <= 48 && lx >= 1 && lx <= 48 &&
            (unsigned)gy < 512u && (unsigned)gx < 512u) {
            v = din[gy * 512 + gx];
            z = d0 [gy * 512 + gx];
        }
        sA[idx] = v;
        sB[idx] = v;
        s0[idx] = z;
    }
    __syncthreads();

    // Precompute this thread's 9 cells (2304 = 9*256, no remainder).
    int   cc[9];
    bool  ig[9];
    float zz[9];
#pragma unroll
    for (int s = 0; s < 9; ++s) {
        int q  = t + 256 * s;
        int ly = q / 48 + 1, lx = q % 48 + 1;
        cc[s] = ly * 50 + lx;
        int gy = gy0 + ly, gx = gx0 + lx;
        ig[s] = ((unsigned)gy < 512u) && ((unsigned)gx < 512u);
        zz[s] = s0[cc[s]];
    }

    float* cur = sA;
    float* nxt = sB;
    for (int it = 0; it < nsweeps; ++it) {
#pragma unroll
        for (int s = 0; s < 9; ++s) {
            if (ig[s]) {
                int c = cc[s];
                float a  = fminf(cur[c - 50], cur[c + 50]);  // vertical neighbors
                float b  = fminf(cur[c - 1],  cur[c + 1]);   // horizontal neighbors
                float mn = fminf(a, b);
                float one  = mn + 1.0f;
                float diff = a - b;
                float two  = 0.5f * (a + b + sqrtf(fmaxf(2.0f - diff * diff, 1e-12f)));
                float nw   = (fabsf(diff) >= 1.0f) ? one : two;
                float dn2  = fminf(cur[c], nw);
                float z    = zz[s];
                nxt[c] = (z < BIGF) ? z : dn2;               // frozen override
            }
        }
        __syncthreads();
        float* tmp = cur; cur = nxt; nxt = tmp;
    }

    // Write back the exact 32x32 interior.
    for (int q = t; q < 1024; q += 256) {
        int iy = q >> 5, ix = q & 31;
        int gy = TI * 32 + iy, gx = TJ * 32 + ix;
        dout[gy * 512 + gx] = cur[(iy + 9) * 50 + (ix + 9)];
    }
}

// ---------------------------------------------------------------------------
// Kernels 3/4: min & max of sd = d*sign(phi), two-stage tree reduce.
// ---------------------------------------------------------------------------
__global__ __launch_bounds__(256) void reduce1_kernel(const float* __restrict__ d,
                                                      const float* __restrict__ phi,
                                                      float* __restrict__ partial) {
    __shared__ float smn[256];
    __shared__ float smx[256];
    int t = threadIdx.x;
    int base = blockIdx.x * 1024;
    float mn = 3.4e38f, mx = -3.4e38f;
#pragma unroll
    for (int s = 0; s < 4; ++s) {
        int i = base + t + 256 * s;
        float p  = phi[i];
        float sd = d[i] * (p >= 0.0f ? 1.0f : -1.0f);
        mn = fminf(mn, sd);
        mx = fmaxf(mx, sd);
    }
    smn[t] = mn; smx[t] = mx;
    __syncthreads();
    for (int off = 128; off > 0; off >>= 1) {
        if (t < off) {
            smn[t] = fminf(smn[t], smn[t + off]);
            smx[t] = fmaxf(smx[t], smx[t + off]);
        }
        __syncthreads();
    }
    if (t == 0) {
        partial[blockIdx.x]       = smn[0];
        partial[256 + blockIdx.x] = smx[0];
    }
}

__global__ __launch_bounds__(256) void reduce2_kernel(const float* __restrict__ partial,
                                                      float* __restrict__ minmax) {
    __shared__ float smn[256];
    __shared__ float smx[256];
    int t = threadIdx.x;
    smn[t] = partial[t];
    smx[t] = partial[256 + t];
    __syncthreads();
    for (int off = 128; off > 0; off >>= 1) {
        if (t < off) {
            smn[t] = fminf(smn[t], smn[t + off]);
            smx[t] = fmaxf(smx[t], smx[t + off]);
        }
        __syncthreads();
    }
    if (t == 0) { minmax[0] = smn[0]; minmax[1] = smx[0]; }
}

// ---------------------------------------------------------------------------
// Kernel 5: 4x bilinear upsample 512x512 -> 2048x2048 via f32 WMMA.
// Out_tile(16x16) = Wr(16x16) * ( S(16x16) * WcT(16x16) ), each 16x16x16
// matmul chained as 4x V_WMMA_F32_16X16X4_F32 (K-chunks of 4).
// Triangle-kernel weight: x = 0.25*out - 0.375 (exact in f32; edge
// renormalization in JAX == index clamping for a 2-tap kernel).
// ---------------------------------------------------------------------------
__device__ __forceinline__ float wfun(int outIdx, int patchIdx, int base) {
    float x  = 0.25f * (float)outIdx - 0.375f;
    float ff = floorf(x);
    int   f  = (int)ff;
    float w  = x - ff;
    int lo = min(max(f, 0), 511) - base;
    int hi = min(max(f + 1, 0), 511) - base;
    float r = 0.0f;
    if (lo == patchIdx) r += 1.0f - w;
    if (hi == patchIdx) r += w;
    return r;
}

__global__ __launch_bounds__(256) void resize_kernel(const float* __restrict__ d,
                                                     const float* __restrict__ phi,
                                                     const float* __restrict__ minmax,
                                                     float* __restrict__ out) {
    int lane = threadIdx.x & 31;
    int wave = threadIdx.x >> 5;
    int tile = blockIdx.x * 8 + wave;          // 16384 tiles of 16x16
    int TI = tile >> 7, TJ = tile & 127;
    int I0 = TI * 16, J0 = TJ * 16;
    int n  = lane & 15;                        // column / row index within 16
    int hh = lane >> 4;                        // half-wave select

    float mn = minmax[0], mx = minmax[1];
    float sc = 2.0f / (mx - mn);

    int R0 = I0 >> 2, C0 = J0 >> 2;
    int rbase = min(max(R0 - 1, 0), 496);      // 16-row input patch base (in range)
    int cbase = min(max(C0 - 1, 0), 496);      // 16-col input patch base (in range)
    int prow  = rbase + n;                     // A-matrix row held by this lane

    // ---- T = S_patch x WcT  (K = patch columns) -------------------------
    v8f acc = {};
#pragma unroll
    for (int k = 0; k < 4; ++k) {
        int pc = cbase + 4 * k + 2 * hh;       // contiguous pair of patch cols
        int gi = prow * 512 + pc;
        float d0v = d[gi],     d1v = d[gi + 1];
        float p0  = phi[gi],   p1  = phi[gi + 1];
        v2f a;
        a.x = ((p0 >= 0.0f ? d0v : -d0v) - mn) * sc - 1.0f;
        a.y = ((p1 >= 0.0f ? d1v : -d1v) - mn) * sc - 1.0f;
        v2f b;
        b.x = wfun(J0 + n, 4 * k + 2 * hh + 0, cbase);
        b.y = wfun(J0 + n, 4 * k + 2 * hh + 1, cbase);
        acc = __builtin_amdgcn_wmma_f32_16x16x4_f32(false, a, false, b,
                                                    (short)0, acc, false, false);
    }

    // ---- Out = Wr x T : relayout T (C/D layout) into B-operand chunks ---
    // row m, col n of T lives in lane n + 16*(m>>3), VGPR m&7.
    v8f o = {};
#pragma unroll
    for (int k = 0; k < 4; ++k) {
        int src = n + ((k & 2) ? 16 : 0);
        int vb  = (k & 1) * 4;                 // constant after unroll
        float l0 = __shfl(acc[vb + 0], src, 32);
        float l1 = __shfl(acc[vb + 1], src, 32);
        float h0 = __shfl(acc[vb + 2], src, 32);
        float h1 = __shfl(acc[vb + 3], src, 32);
        v2f b;
        b.x = hh ? h0 : l0;
        b.y = hh ? h1 : l1;
        v2f a;
        a.x = wfun(I0 + n, 4 * k + 2 * hh + 0, rbase);
        a.y = wfun(I0 + n, 4 * k + 2 * hh + 1, rbase);
        o = __builtin_amdgcn_wmma_f32_16x16x4_f32(false, a, false, b,
                                                  (short)0, o, false, false);
    }

    // ---- store D (lane: col n; VGPR v: row v + 8*hh) --------------------
#pragma unroll
    for (int v = 0; v < 8; ++v) {
        int row = I0 + v + 8 * hh;
        out[row * 2048 + J0 + n] = o[v];
    }
}

// ---------------------------------------------------------------------------
// Host launcher
// ---------------------------------------------------------------------------
extern "C" void kernel_launch(void* const* d_in, const int* in_sizes, int n_in,
                              void* d_out, int out_size, void* d_ws, size_t ws_size,
                              hipStream_t stream) {
    (void)in_sizes; (void)n_in; (void)out_size; (void)ws_size;
    const int N = 512 * 512;

    const float* phi = (const float*)d_in[0];
    float* ws      = (float*)d_ws;
    float* d0      = ws;                 // N floats
    float* dA      = ws + N;             // N floats
    float* dB      = ws + 2 * N;         // N floats
    float* partial = ws + 3 * N;         // 512 floats
    float* minmax  = partial + 512;      // 2 floats
    float* out     = (float*)d_out;      // 2048*2048 floats

    init_kernel<<<N / 256, 256, 0, stream>>>(phi, d0, dA);

    // 37 launches x 8 sweeps + 1 launch x 4 sweeps = 300 sweeps total.
    float* cin  = dA;
    float* cout = dB;
    for (int l = 0; l < 38; ++l) {
        int T = (l == 37) ? 4 : 8;
        sweep_kernel<<<dim3(16, 16, 1), 256, 0, stream>>>(cin, cout, d0, T);
        float* tmp = cin; cin = cout; cout = tmp;
    }
    // final distance field is in `cin`

    reduce1_kernel<<<256, 256, 0, stream>>>(cin, phi, partial);
    reduce2_kernel<<<1, 256, 0, stream>>>(partial, minmax);

    resize_kernel<<<2048, 256, 0, stream>>>(cin, phi, minmax, out);
}